// GP_EP_CondRegLayer_31748398252615
// MI455X (gfx1250) — compile-verified
//
#include <hip/hip_runtime.h>
#include <hip/hip_bf16.h>

typedef __attribute__((ext_vector_type(2))) float v2f;
typedef __attribute__((ext_vector_type(8))) float v8f;

#define NBS   2048          // K*M = 16*128 batch (b,s) pairs
#define DD    16            // D dimension (per-tau)
#define TIN   302           // input T
#define TOUT  308           // output T
#define EPSJ  1e-4f

// ---------------------------------------------------------------------------
// Kernel 1: per (j,d) build W = [A_mu | L] in ws and per-block ldj partials.
// Grid: 48 blocks (j*16+d), 256 threads.
// ---------------------------------------------------------------------------
__global__ __launch_bounds__(256) void setup_chol_kernel(
    const float* __restrict__ ts,        // 150 floats (flattened 3x50)
    const float* __restrict__ log_tau,   // 16 floats
    float* __restrict__ ws,              // W arrays base
    float* __restrict__ ldj_partial)     // 48 floats
{
  const int blk = blockIdx.x;
  const int j   = blk >> 4;
  const int d   = blk & 15;
  const int m   = 50 * (j + 1);
  const int C   = (m + 2 + 3) & ~3;        // 52,104,152
  const int R   = (m + 15) & ~15;          // 64,112,160
  const float tpsj = 10.0f * (float)(j + 1);

  int off = 0;
  if (j >= 1) off += 16 * 64 * 52;
  if (j >= 2) off += 16 * 112 * 104;
  float* W = ws + off + d * (R * C);

  const int tid = threadIdx.x;
  const int nt  = blockDim.x;

  __shared__ float tsj[160];
  __shared__ float colk[160];
  __shared__ float invd;
  __shared__ float red[256];

  for (int i = tid; i < m; i += nt) tsj[i] = ts[i];
  __syncthreads();

  const float tau = __builtin_expf(log_tau[d]);
  const float a2  = 1.0f / (2.0f * tau * tau);
  // 2x2 K_EPEP = [[1, k01],[k01, 1]]  ->  inverse
  const float k01 = __builtin_expf(-(tpsj * tpsj) * a2);
  const float det = 1.0f - k01 * k01;
  const float i00 = 1.0f / det;
  const float i01 = -k01 / det;

  // Fill W: cols [0,2) = A_mu, cols [2, 2+m) = Sigma (pre-Cholesky), rest 0.
  for (int p = tid; p < R; p += nt) {
    float* row = W + p * C;
    if (p < m) {
      const float tp  = tsj[p];
      const float kp0 = __builtin_expf(-(tp * tp) * a2);
      const float e1  = tp - tpsj;
      const float kp1 = __builtin_expf(-(e1 * e1) * a2);
      const float A0  = kp0 * i00 + kp1 * i01;
      const float A1  = kp0 * i01 + kp1 * i00;
      row[0] = A0;
      row[1] = A1;
      for (int q = 0; q < m; ++q) {
        const float tq  = tsj[q];
        const float dt  = tp - tq;
        const float kq0 = __builtin_expf(-(tq * tq) * a2);
        const float f1  = tq - tpsj;
        const float kq1 = __builtin_expf(-(f1 * f1) * a2);
        float s = __builtin_expf(-(dt * dt) * a2) - A0 * kq0 - A1 * kq1;
        if (q == p) s += EPSJ;
        row[2 + q] = s;
      }
      for (int q = m + 2; q < C; ++q) row[q] = 0.0f;
    } else {
      for (int q = 0; q < C; ++q) row[q] = 0.0f;
    }
  }
  __syncthreads();

  // In-place right-looking Cholesky on W[:, 2:2+m], pivot column in LDS.
  for (int k = 0; k < m; ++k) {
    if (tid == 0) {
      const float v = __builtin_sqrtf(W[k * C + 2 + k]);
      W[k * C + 2 + k] = v;
      invd = 1.0f / v;
    }
    __syncthreads();
    for (int p = k + 1 + tid; p < m; p += nt) {
      const float v = W[p * C + 2 + k] * invd;
      W[p * C + 2 + k] = v;
      colk[p] = v;
    }
    __syncthreads();
    for (int p = k + 1 + tid; p < m; p += nt) {
      const float cp = colk[p];
      float* row = W + p * C + 2;
      for (int q = k + 1; q <= p; ++q) row[q] -= cp * colk[q];
    }
    __syncthreads();
  }

  // Zero strict upper triangle (GEMM multiplies the full row).
  for (int p = tid; p < m; p += nt) {
    float* row = W + p * C + 2;
    for (int q = p + 1; q < m; ++q) row[q] = 0.0f;
  }

  // ldj partial: sum log|diag| with fixed-order tree reduction.
  float s = 0.0f;
  for (int k = tid; k < m; k += nt)
    s += __builtin_logf(__builtin_fabsf(W[k * C + 2 + k]));
  red[tid] = s;
  __syncthreads();
  for (int w = nt >> 1; w > 0; w >>= 1) {
    if (tid < w) red[tid] += red[tid + w];
    __syncthreads();
  }
  if (tid == 0) ldj_partial[blk] = red[0];
}

// ---------------------------------------------------------------------------
// Kernel 2: pass-through endpoint columns + deterministic ldj finalize.
// ---------------------------------------------------------------------------
__global__ __launch_bounds__(256) void copy_ep_kernel(
    const float* __restrict__ z,
    const float* __restrict__ sldj,
    const float* __restrict__ ldj_partial,
    float* __restrict__ out)
{
  const int idx = blockIdx.x * blockDim.x + threadIdx.x;   // over NBS*DD rows
  if (idx < NBS * DD) {
    const float* zr = z + (size_t)idx * TIN;
    float* orow     = out + (size_t)idx * TOUT;
    const float z0 = zr[0], z1 = zr[1];
    orow[0]   = z0; orow[51]  = z1;    // block j=0: [0,52)
    orow[52]  = z0; orow[153] = z1;    // block j=1: [52,154)
    orow[154] = z0; orow[307] = z1;    // block j=2: [154,308)
  }
  if (idx == 0) {
    float s = sldj[0];
    for (int i = 0; i < 48; ++i) s += ldj_partial[i];
    out[(size_t)NBS * DD * TOUT] = s;
  }
}

// ---------------------------------------------------------------------------
// Kernel 3: Out[n,p] = sum_k X[n,k] * W[p,k] via V_WMMA_F32_16X16X4_F32.
//   X[n,0:2] = z[n,d,0:2] (endpoints), X[n,2+q] = z[n,d,t0+q] (segment).
// One workgroup per (d, p-tile); W tile staged in LDS; 8 waves sweep n-tiles.
// ---------------------------------------------------------------------------
__global__ __launch_bounds__(256) void gemm_wmma_kernel(
    const float* __restrict__ z,
    const float* __restrict__ Wall,
    float* __restrict__ out,
    int m, int C, int nptiles, int t0_in, int t_out_base, int woff)
{
  alignas(16) __shared__ float Wt[16 * 152];

  const int d   = blockIdx.x / nptiles;
  const int pt  = blockIdx.x % nptiles;
  const int tid = threadIdx.x;

  const float* Wsrc = Wall + woff + (size_t)d * (nptiles * 16) * C + (size_t)pt * 16 * C;
  for (int i = tid; i < 16 * C; i += 256) Wt[i] = Wsrc[i];
  __syncthreads();

  const int lane = tid & 31;
  const int wave = tid >> 5;
  const int lo   = lane & 15;
  const int hi   = lane >> 4;

  const int p_g = pt * 16 + lo;               // B column / output t row index

  for (int ntile = wave; ntile < NBS / 16; ntile += 8) {
    const int n_row = ntile * 16 + lo;        // A row (M dim)
    const float* zrow = z + ((size_t)n_row * DD + d) * TIN;

    v8f acc = {};
    for (int k0 = 0; k0 < C; k0 += 4) {
      const int ka = k0 + 2 * hi;             // even; pair (ka, ka+1) contiguous
      // A (16x4 f32): lane holds X[n_row, ka], X[n_row, ka+1]
      const float* ap = (ka == 0) ? zrow : (zrow + t0_in + (ka - 2));
      v2f a = *(const v2f*)ap;
      // B (4x16 f32): lane holds W[p_g, ka], W[p_g, ka+1]  (B[k][p] = W[p][k])
      v2f b = *(const v2f*)(&Wt[lo * C + ka]);
      acc = __builtin_amdgcn_wmma_f32_16x16x4_f32(
          /*neg_a=*/false, a, /*neg_b=*/false, b,
          /*c_mod=*/(short)0, acc, /*reuse_a=*/false, /*reuse_b=*/false);
    }

    if (p_g < m) {
      const int t = t_out_base + 1 + p_g;
#pragma unroll
      for (int v = 0; v < 8; ++v) {
        const int n_g = ntile * 16 + v + 8 * hi;   // C/D: M = v + 8*hi, N = lane&15
        out[((size_t)n_g * DD + d) * TOUT + t] = acc[v];
      }
    }
  }
}

// ---------------------------------------------------------------------------
extern "C" void kernel_launch(void* const* d_in, const int* in_sizes, int n_in,
                              void* d_out, int out_size, void* d_ws, size_t ws_size,
                              hipStream_t stream) {
  const float* z       = (const float*)d_in[0];   // 16*128*16*302
  const float* ts      = (const float*)d_in[1];   // 3*50
  const float* log_tau = (const float*)d_in[2];   // 16
  const float* sldj    = (const float*)d_in[3];   // 1
  float* out = (float*)d_out;
  float* ws  = (float*)d_ws;

  const int offW[3] = {0, 16 * 64 * 52, 16 * 64 * 52 + 16 * 112 * 104};
  const int ldjOff  = offW[2] + 16 * 160 * 152;   // ~628K floats of W + 48 partials

  setup_chol_kernel<<<48, 256, 0, stream>>>(ts, log_tau, ws, ws + ldjOff);
  copy_ep_kernel<<<(NBS * DD + 255) / 256, 256, 0, stream>>>(z, sldj, ws + ldjOff, out);

  const int ms[3]  = {50, 100, 150};
  const int Cs[3]  = {52, 104, 152};
  const int npt[3] = {4, 7, 10};
  const int t0s[3] = {2, 52, 152};
  const int tob[3] = {0, 52, 154};
  for (int j = 0; j < 3; ++j) {
    gemm_wmma_kernel<<<16 * npt[j], 256, 0, stream>>>(
        z, ws, out, ms[j], Cs[j], npt[j], t0s[j], tob[j], offW[j]);
  }
}